// SlotInitSAMLPMax_40183714021834
// MI455X (gfx1250) — compile-verified
//
#include <hip/hip_runtime.h>
#include <hip/hip_bf16.h>
#include <math.h>

// ---------------------------------------------------------------------------
// SlotInitSAMLPMax for MI455X (gfx1250, wave32)
// Phase 1: tiny iterative slot-attention update on cc[10][64] - one workgroup.
// Phase 2: per-slot MLP+max, HBM-bound (134MB of Wa/Wb), f32 WMMA 16x16x4,
//          software-pipelined B-fragment loads, async global->LDS cc staging.
// ---------------------------------------------------------------------------

#define NUM_SLOTS   4096
#define NUM_CLUSTER 10
#define D           64

typedef __attribute__((ext_vector_type(2))) float v2f;
typedef __attribute__((ext_vector_type(8))) float v8f;

// ============================ Phase 1 ======================================
// One block, 640 threads: thread (n,i) with n = tid>>6 (cluster), i = tid&63.
__global__ void __launch_bounds__(640)
slot_phase1_kernel(const float* __restrict__ cc0,
                   const float* __restrict__ Wk, const float* __restrict__ bk,
                   const float* __restrict__ Wq, const float* __restrict__ bq,
                   const float* __restrict__ Wv, const float* __restrict__ bv,
                   const float* __restrict__ cc_g, const float* __restrict__ cc_b,
                   const float* __restrict__ W_ih, const float* __restrict__ W_hh,
                   const float* __restrict__ b_ih, const float* __restrict__ b_hh,
                   const float* __restrict__ ln_g, const float* __restrict__ ln_b,
                   const float* __restrict__ W1, const float* __restrict__ b1,
                   const float* __restrict__ W2, const float* __restrict__ b2,
                   float* __restrict__ cc_out)
{
    __shared__ float cc[NUM_CLUSTER][D];
    __shared__ float kk[NUM_CLUSTER][D];
    __shared__ float vv[NUM_CLUSTER][D];
    __shared__ float xx[NUM_CLUSTER][D];
    __shared__ float qq[NUM_CLUSTER][D];
    __shared__ float upd[NUM_CLUSTER][D];
    __shared__ float tmp[NUM_CLUSTER][D];
    __shared__ float h2[NUM_CLUSTER][2 * D];
    __shared__ float attn[NUM_CLUSTER][NUM_CLUSTER];
    __shared__ float sm[NUM_CLUSTER][NUM_CLUSTER];
    __shared__ float mean_[NUM_CLUSTER];
    __shared__ float rstd_[NUM_CLUSTER];

    const int tid = threadIdx.x;
    const int n = tid >> 6;
    const int i = tid & 63;
    const float temp = 0.125f;  // 64^-0.5

    cc[n][i] = cc0[tid];
    __syncthreads();

    // k = cc @ Wk^T + bk ; v = cc @ Wv^T + bv
    {
        float sk = bk[i], sv = bv[i];
        for (int j = 0; j < D; ++j) {
            float c = cc[n][j];
            sk += c * Wk[i * D + j];
            sv += c * Wv[i * D + j];
        }
        kk[n][i] = sk;
        vv[n][i] = sv;
    }
    __syncthreads();

    for (int it = 0; it < 3; ++it) {
        // ---- LayerNorm(cc) with cc_g/cc_b ----
        if (tid < NUM_CLUSTER) {
            float m = 0.f;
            for (int j = 0; j < D; ++j) m += cc[tid][j];
            m *= (1.f / D);
            float vs = 0.f;
            for (int j = 0; j < D; ++j) { float d0 = cc[tid][j] - m; vs += d0 * d0; }
            mean_[tid] = m;
            rstd_[tid] = rsqrtf(vs * (1.f / D) + 1e-5f);
        }
        __syncthreads();
        xx[n][i] = (cc[n][i] - mean_[n]) * rstd_[n] * cc_g[i] + cc_b[i];
        __syncthreads();

        // q = ln(cc) @ Wq^T + bq
        {
            float s = bq[i];
            for (int j = 0; j < D; ++j) s += xx[n][j] * Wq[i * D + j];
            qq[n][i] = s;
        }
        __syncthreads();

        // attn logits [n][m] = k[n].q[m] * temp
        if (tid < NUM_CLUSTER * NUM_CLUSTER) {
            int an = tid / NUM_CLUSTER, am = tid % NUM_CLUSTER;
            float s = 0.f;
            for (int j = 0; j < D; ++j) s += kk[an][j] * qq[am][j];
            attn[an][am] = s * temp;
        }
        __syncthreads();

        // softmax over axis 0 (over n, per column m), then + EPS
        if (tid < NUM_CLUSTER) {
            int m = tid;
            float mx = -1e30f;
            for (int nn = 0; nn < NUM_CLUSTER; ++nn) mx = fmaxf(mx, attn[nn][m]);
            float s = 0.f;
            for (int nn = 0; nn < NUM_CLUSTER; ++nn) {
                float e = __expf(attn[nn][m] - mx);
                sm[nn][m] = e;
                s += e;
            }
            float inv = 1.f / s;
            for (int nn = 0; nn < NUM_CLUSTER; ++nn) sm[nn][m] = sm[nn][m] * inv + 1e-8f;
        }
        __syncthreads();
        // renormalize per row n
        if (tid < NUM_CLUSTER) {
            int nn = tid;
            float s = 0.f;
            for (int m = 0; m < NUM_CLUSTER; ++m) s += sm[nn][m];
            float inv = 1.f / s;
            for (int m = 0; m < NUM_CLUSTER; ++m) attn[nn][m] = sm[nn][m] * inv;
        }
        __syncthreads();

        // updates = attn @ v
        {
            float s = 0.f;
            for (int m = 0; m < NUM_CLUSTER; ++m) s += attn[n][m] * vv[m][i];
            upd[n][i] = s;
        }
        __syncthreads();

        // ---- GRUCell ----
        {
            float ir = b_ih[i], iz = b_ih[i + D], in_ = b_ih[i + 2 * D];
            float hr = b_hh[i], hz = b_hh[i + D], hn = b_hh[i + 2 * D];
            for (int j = 0; j < D; ++j) {
                float u = upd[n][j], c = cc[n][j];
                ir  += u * W_ih[(i)         * D + j];
                iz  += u * W_ih[(i + D)     * D + j];
                in_ += u * W_ih[(i + 2 * D) * D + j];
                hr  += c * W_hh[(i)         * D + j];
                hz  += c * W_hh[(i + D)     * D + j];
                hn  += c * W_hh[(i + 2 * D) * D + j];
            }
            float r  = 1.f / (1.f + __expf(-(ir + hr)));
            float z  = 1.f / (1.f + __expf(-(iz + hz)));
            float nv = tanhf(in_ + r * hn);
            tmp[n][i] = (1.f - z) * nv + z * cc[n][i];
        }
        __syncthreads();
        cc[n][i] = tmp[n][i];
        __syncthreads();

        // ---- residual MLP: cc += relu(ln(cc)@W1^T+b1)@W2^T + b2 ----
        if (tid < NUM_CLUSTER) {
            float m = 0.f;
            for (int j = 0; j < D; ++j) m += cc[tid][j];
            m *= (1.f / D);
            float vs = 0.f;
            for (int j = 0; j < D; ++j) { float d0 = cc[tid][j] - m; vs += d0 * d0; }
            mean_[tid] = m;
            rstd_[tid] = rsqrtf(vs * (1.f / D) + 1e-5f);
        }
        __syncthreads();
        xx[n][i] = (cc[n][i] - mean_[n]) * rstd_[n] * ln_g[i] + ln_b[i];
        __syncthreads();
        {
            float s1 = b1[i], s2 = b1[i + D];
            for (int j = 0; j < D; ++j) {
                float x = xx[n][j];
                s1 += x * W1[i * D + j];
                s2 += x * W1[(i + D) * D + j];
            }
            h2[n][i]     = fmaxf(s1, 0.f);
            h2[n][i + D] = fmaxf(s2, 0.f);
        }
        __syncthreads();
        {
            float s = b2[i];
            for (int j = 0; j < 2 * D; ++j) s += h2[n][j] * W2[i * 2 * D + j];
            tmp[n][i] = cc[n][i] + s;
        }
        __syncthreads();
        cc[n][i] = tmp[n][i];
        __syncthreads();
    }

    cc_out[tid] = cc[n][i];
}

// ============================ Phase 2 ======================================
// One wave per slot. H = CCpad(16x64) @ Wa[s]^T, OUT = H @ Wb[s]^T; max rows.
// f32 WMMA 16x16x4: A/B fragments = 2 VGPRs each.
//   A (16x4): lane l<16 -> {A[l][k0], A[l][k0+1]}, lane l>=16 -> {A[l-16][k0+2], A[l-16][k0+3]}
//   B (4x16): lane l<16 -> {B[k0][l], B[k0+1][l]}, lane l>=16 -> {B[k0+2][l-16], B[k0+3][l-16]}
//   D (16x16): lane l, reg r -> D[r + 8*(l>=16)][l&15]

__device__ __forceinline__ void load_btile(const float* __restrict__ base, int t,
                                           int l15, int half, v2f bf[16]) {
    const float* wrow = base + (size_t)(16 * t + l15) * D + 2 * half;
#pragma unroll
    for (int c = 0; c < 16; ++c)
        bf[c] = *(const v2f*)(wrow + 4 * c);
}

__global__ void __launch_bounds__(128)
slot_phase2_kernel(const float* __restrict__ cc,   // [10][64] from phase 1
                   const float* __restrict__ Wa, const float* __restrict__ ba,
                   const float* __restrict__ Wb, const float* __restrict__ bb,
                   float* __restrict__ out)         // [4096][64]
{
    __shared__ float ccS[16][D];     // zero-padded cc tile (4 KB), shared by 4 waves
    __shared__ float Hs[4][16][D];   // per-wave H tile (4 KB each)

    const int tid  = threadIdx.x;
    const int lane = tid & 31;
    const int wid  = tid >> 5;
    const int slot = blockIdx.x * 4 + wid;
    const int half = lane >> 4;      // 0: lanes 0-15, 1: lanes 16-31
    const int l15  = lane & 15;

    const float* __restrict__ WaS = Wa + (size_t)slot * D * D;
    const float* __restrict__ WbS = Wb + (size_t)slot * D * D;
    const float* __restrict__ baS = ba + (size_t)slot * D;
    const float* __restrict__ bbS = bb + (size_t)slot * D;
    float* __restrict__ outS = out + (size_t)slot * D;

    // ---- Stage cc (2560B) into LDS via CDNA5 async global->LDS b128 copies.
    // 160 granules of 16B; threads 0..127 then 0..31 (ASYNCcnt-tracked).
    {
        const int ngran = NUM_CLUSTER * D * 4 / 16;  // 160
        for (int g = tid; g < ngran; g += 128) {
            unsigned long long ga = (unsigned long long)(const void*)cc + (size_t)g * 16;
            unsigned int la = (unsigned int)(unsigned long long)(const void*)&ccS[0][0]
                              + (unsigned int)(g * 16);
            asm volatile("global_load_async_to_lds_b128 %0, %1, off"
                         :: "v"(la), "v"(ga) : "memory");
        }
        // zero the padding rows 10..15 (384 floats)
        for (int g = tid; g < (16 - NUM_CLUSTER) * D; g += 128)
            ccS[NUM_CLUSTER + (g >> 6)][g & 63] = 0.f;
        asm volatile("s_wait_asynccnt 0x0" ::: "memory");
    }
    __syncthreads();

    // Prefetch Wb[s] (16KB = 128 x 128B lines) while GEMM1 streams Wa[s].
#pragma unroll
    for (int p = 0; p < 4; ++p)
        __builtin_prefetch(WbS + (size_t)(p * 32 + lane) * 32, 0, 1);

    // A-fragments of zero-padded CC from LDS (no divergence).
    v2f af[16];
#pragma unroll
    for (int c = 0; c < 16; ++c) {
        int col = 4 * c + 2 * half;
        v2f a;
        a.x = ccS[l15][col];
        a.y = ccS[l15][col + 1];
        af[c] = a;
    }

    // ---- GEMM1: H = relu(CCpad @ Wa^T + ba) -> LDS ----
    // Software pipeline: preload tile t+1 B-fragments while tile t does WMMA.
    v2f bf[16];
    load_btile(WaS, 0, l15, half, bf);
#pragma unroll
    for (int t = 0; t < 4; ++t) {
        v2f bn[16];
        if (t < 3) load_btile(WaS, t + 1, l15, half, bn);
        else       load_btile(WbS, 0,     l15, half, bn);  // bridge into GEMM2

        v8f acc0 = {}, acc1 = {};
#pragma unroll
        for (int c = 0; c < 16; c += 2) {
            acc0 = __builtin_amdgcn_wmma_f32_16x16x4_f32(
                false, af[c],     false, bf[c],     (short)0, acc0, false, false);
            acc1 = __builtin_amdgcn_wmma_f32_16x16x4_f32(
                false, af[c + 1], false, bf[c + 1], (short)0, acc1, false, false);
        }
        v8f acc = acc0 + acc1;

        float bias = baS[16 * t + l15];
#pragma unroll
        for (int r = 0; r < 8; ++r) {
            int m = r + 8 * half;
            Hs[wid][m][16 * t + l15] = fmaxf(acc[r] + bias, 0.f);
        }
#pragma unroll
        for (int c = 0; c < 16; ++c) bf[c] = bn[c];
    }

    // Same-wave LDS RAW: make sure all DS stores completed before re-reading.
    asm volatile("s_wait_dscnt 0x0" ::: "memory");

    // A-fragments of H from LDS (overlaps the in-flight Wb tile-0 loads).
    v2f hf[16];
#pragma unroll
    for (int c = 0; c < 16; ++c) {
        int col = 4 * c + 2 * half;
        v2f a;
        a.x = Hs[wid][l15][col];
        a.y = Hs[wid][l15][col + 1];
        hf[c] = a;
    }

    // ---- GEMM2: OUT = H @ Wb^T + bb, then max over rows n<10 ----
#pragma unroll
    for (int t = 0; t < 4; ++t) {
        v2f bn[16];
        if (t < 3) load_btile(WbS, t + 1, l15, half, bn);

        v8f acc0 = {}, acc1 = {};
#pragma unroll
        for (int c = 0; c < 16; c += 2) {
            acc0 = __builtin_amdgcn_wmma_f32_16x16x4_f32(
                false, hf[c],     false, bf[c],     (short)0, acc0, false, false);
            acc1 = __builtin_amdgcn_wmma_f32_16x16x4_f32(
                false, hf[c + 1], false, bf[c + 1], (short)0, acc1, false, false);
        }
        v8f acc = acc0 + acc1;

        float bias = bbS[16 * t + l15];
        float mx = -INFINITY;
#pragma unroll
        for (int r = 0; r < 8; ++r) {
            int m = r + 8 * half;
            float vv_ = acc[r] + bias;
            if (m < NUM_CLUSTER) mx = fmaxf(mx, vv_);
        }
        // Combine the two half-wave partial maxima (same output column i).
        float other = __shfl_xor(mx, 16, 32);
        mx = fmaxf(mx, other);
        if (half == 0) outS[16 * t + l15] = mx;

#pragma unroll
        for (int c = 0; c < 16; ++c) bf[c] = bn[c];
    }
}

// ============================ Launch =======================================
extern "C" void kernel_launch(void* const* d_in, const int* in_sizes, int n_in,
                              void* d_out, int out_size, void* d_ws, size_t ws_size,
                              hipStream_t stream) {
    const float* cc0   = (const float*)d_in[0];
    const float* Wk    = (const float*)d_in[1];
    const float* bk    = (const float*)d_in[2];
    const float* Wq    = (const float*)d_in[3];
    const float* bq    = (const float*)d_in[4];
    const float* Wv    = (const float*)d_in[5];
    const float* bv    = (const float*)d_in[6];
    const float* cc_g  = (const float*)d_in[7];
    const float* cc_b  = (const float*)d_in[8];
    const float* W_ih  = (const float*)d_in[9];
    const float* W_hh  = (const float*)d_in[10];
    const float* b_ih  = (const float*)d_in[11];
    const float* b_hh  = (const float*)d_in[12];
    const float* ln_g  = (const float*)d_in[13];
    const float* ln_b  = (const float*)d_in[14];
    const float* W1    = (const float*)d_in[15];
    const float* b1    = (const float*)d_in[16];
    const float* W2    = (const float*)d_in[17];
    const float* b2    = (const float*)d_in[18];
    const float* Wa    = (const float*)d_in[19];
    const float* ba    = (const float*)d_in[20];
    const float* Wb    = (const float*)d_in[21];
    const float* bb    = (const float*)d_in[22];

    float* cc_final = (float*)d_ws;   // [10][64]

    slot_phase1_kernel<<<1, NUM_CLUSTER * D, 0, stream>>>(
        cc0, Wk, bk, Wq, bq, Wv, bv, cc_g, cc_b,
        W_ih, W_hh, b_ih, b_hh, ln_g, ln_b, W1, b1, W2, b2, cc_final);

    slot_phase2_kernel<<<NUM_SLOTS / 4, 128, 0, stream>>>(
        cc_final, Wa, ba, Wb, bb, (float*)d_out);
}